// DualStaticVolume_30545807409279
// MI455X (gfx1250) — compile-verified
//
#include <hip/hip_runtime.h>
#include <math.h>

#define EPSF 1e-8f
#define NSAMP 64

typedef __attribute__((ext_vector_type(2))) float v2f;
typedef __attribute__((ext_vector_type(8))) float v8f;

// Raw LDS byte offset of a generic pointer into __shared__ (LDS aperture keeps
// the offset in the low 32 bits per ISA 10.2).
__device__ __forceinline__ unsigned lds_off(void* p) {
  return (unsigned)(unsigned long long)(__attribute__((address_space(3))) void*)p;
}

// Classic adjugate 4x4 inverse (layout-agnostic: row-major in -> row-major out).
__device__ void inv4x4(const float* m, float* o) {
  float i0  =  m[5]*m[10]*m[15] - m[5]*m[11]*m[14] - m[9]*m[6]*m[15] + m[9]*m[7]*m[14] + m[13]*m[6]*m[11] - m[13]*m[7]*m[10];
  float i4  = -m[4]*m[10]*m[15] + m[4]*m[11]*m[14] + m[8]*m[6]*m[15] - m[8]*m[7]*m[14] - m[12]*m[6]*m[11] + m[12]*m[7]*m[10];
  float i8  =  m[4]*m[9]*m[15]  - m[4]*m[11]*m[13] - m[8]*m[5]*m[15] + m[8]*m[7]*m[13] + m[12]*m[5]*m[11] - m[12]*m[7]*m[9];
  float i12 = -m[4]*m[9]*m[14]  + m[4]*m[10]*m[13] + m[8]*m[5]*m[14] - m[8]*m[6]*m[13] - m[12]*m[5]*m[10] + m[12]*m[6]*m[9];
  float i1  = -m[1]*m[10]*m[15] + m[1]*m[11]*m[14] + m[9]*m[2]*m[15] - m[9]*m[3]*m[14] - m[13]*m[2]*m[11] + m[13]*m[3]*m[10];
  float i5  =  m[0]*m[10]*m[15] - m[0]*m[11]*m[14] - m[8]*m[2]*m[15] + m[8]*m[3]*m[14] + m[12]*m[2]*m[11] - m[12]*m[3]*m[10];
  float i9  = -m[0]*m[9]*m[15]  + m[0]*m[11]*m[13] + m[8]*m[1]*m[15] - m[8]*m[3]*m[13] - m[12]*m[1]*m[11] + m[12]*m[3]*m[9];
  float i13 =  m[0]*m[9]*m[14]  - m[0]*m[10]*m[13] - m[8]*m[1]*m[14] + m[8]*m[2]*m[13] + m[12]*m[1]*m[10] - m[12]*m[2]*m[9];
  float i2  =  m[1]*m[6]*m[15]  - m[1]*m[7]*m[14]  - m[5]*m[2]*m[15] + m[5]*m[3]*m[14] + m[13]*m[2]*m[7]  - m[13]*m[3]*m[6];
  float i6  = -m[0]*m[6]*m[15]  + m[0]*m[7]*m[14]  + m[4]*m[2]*m[15] - m[4]*m[3]*m[14] - m[12]*m[2]*m[7]  + m[12]*m[3]*m[6];
  float i10 =  m[0]*m[5]*m[15]  - m[0]*m[7]*m[13]  - m[4]*m[1]*m[15] + m[4]*m[3]*m[13] + m[12]*m[1]*m[7]  - m[12]*m[3]*m[5];
  float i14 = -m[0]*m[5]*m[14]  + m[0]*m[6]*m[13]  + m[4]*m[1]*m[14] - m[4]*m[2]*m[13] - m[12]*m[1]*m[6]  + m[12]*m[2]*m[5];
  float i3  = -m[1]*m[6]*m[11]  + m[1]*m[7]*m[10]  + m[5]*m[2]*m[11] - m[5]*m[3]*m[10] - m[9]*m[2]*m[7]   + m[9]*m[3]*m[6];
  float i7  =  m[0]*m[6]*m[11]  - m[0]*m[7]*m[10]  - m[4]*m[2]*m[11] + m[4]*m[3]*m[10] + m[8]*m[2]*m[7]   - m[8]*m[3]*m[6];
  float i11 = -m[0]*m[5]*m[11]  + m[0]*m[7]*m[9]   + m[4]*m[1]*m[11] - m[4]*m[3]*m[9]  - m[8]*m[1]*m[7]   + m[8]*m[3]*m[5];
  float i15 =  m[0]*m[5]*m[10]  - m[0]*m[6]*m[9]   - m[4]*m[1]*m[10] + m[4]*m[2]*m[9]  + m[8]*m[1]*m[6]   - m[8]*m[2]*m[5];
  float det = m[0]*i0 + m[1]*i4 + m[2]*i8 + m[3]*i12;
  float r = 1.0f / det;
  o[0]=i0*r;  o[1]=i1*r;  o[2]=i2*r;  o[3]=i3*r;
  o[4]=i4*r;  o[5]=i5*r;  o[6]=i6*r;  o[7]=i7*r;
  o[8]=i8*r;  o[9]=i9*r;  o[10]=i10*r; o[11]=i11*r;
  o[12]=i12*r; o[13]=i13*r; o[14]=i14*r; o[15]=i15*r;
}

// One-wave setup: ws[0..15] = proj_inv @ view_inv (row-vector convention),
// ws[16..31] = view_inv (needed for mode==0).
__global__ void dsv_setup_kernel(const float* __restrict__ wvt,
                                 const float* __restrict__ proj,
                                 float* __restrict__ ws) {
  if (threadIdx.x != 0) return;
  float vm[16], pm[16], vi[16], pi[16];
  #pragma unroll
  for (int i = 0; i < 16; ++i) { vm[i] = wvt[i]; pm[i] = proj[i]; }
  inv4x4(vm, vi);
  inv4x4(pm, pi);
  #pragma unroll
  for (int i = 0; i < 4; ++i)
    #pragma unroll
    for (int j = 0; j < 4; ++j) {
      float s = 0.f;
      #pragma unroll
      for (int k = 0; k < 4; ++k) s += pi[i*4+k] * vi[k*4+j];
      ws[i*4+j] = s;
    }
  #pragma unroll
  for (int i = 0; i < 16; ++i) ws[16+i] = vi[i];
}

__launch_bounds__(256)
__global__ void dsv_render_kernel(const float* __restrict__ vol,
                                  const int* __restrict__ Hp,
                                  const int* __restrict__ Wp,
                                  const int* __restrict__ modep,
                                  const float* __restrict__ cam,
                                  const float* __restrict__ bbox,
                                  const float* __restrict__ ws,
                                  float* __restrict__ out, int R) {
  // smem[0..15]=combined matrix C, smem[16..31]=view_inv, then 8 waves * 128
  // floats of WMMA extraction scratch.
  __shared__ __align__(16) float smem[32 + 8 * 128];
  const int tid = threadIdx.x;

  // ---- CDNA5 async path: stage the 32-dword matrix block into LDS ----------
  if (tid < 32) {
    unsigned la = lds_off(&smem[tid]);
    unsigned go = (unsigned)(tid * 4);
    asm volatile("global_load_async_to_lds_b32 %0, %1, %2 offset:0"
                 :: "v"(la), "v"(go), "s"(ws) : "memory");
  }
  asm volatile("s_wait_asynccnt 0" ::: "memory");
  __syncthreads();

  const int W = Wp[0], H = Hp[0], mode = modep[0];
  const int lane = tid & 31;
  const int px = blockIdx.x * 16 + (tid & 15);
  const int py = blockIdx.y * 16 + (tid >> 4);

  const float bminx = bbox[0], bminy = bbox[1], bminz = bbox[2];
  const float bmaxx = bbox[3], bmaxy = bbox[4], bmaxz = bbox[5];

  const float ndcx = (W > 1 ? (float)px / (float)(W - 1) : 0.f) * 2.f - 1.f;
  const float ndcy = (H > 1 ? (float)py / (float)(H - 1) : 0.f) * 2.f - 1.f;

  float ox, oy, oz, dx, dy, dz;
  if (mode != 0) {
    // ---- WMMA 16x16x4 f32: world4 = ndc4 @ C for 32 pixels per wave -------
    // A (16x4, ISA layout): lanes 0-15 hold K=0,1 for M=lane; lanes 16-31
    // hold K=2,3 (both == 1.0 since ndc4 = [x,y,1,1]).
    const float nx2 = __shfl(ndcx, (lane & 15) + 16, 32);
    const float ny2 = __shfl(ndcy, (lane & 15) + 16, 32);
    v2f a0, a1, b;
    if (lane < 16) { a0.x = ndcx; a0.y = ndcy; a1.x = nx2; a1.y = ny2; }
    else           { a0.x = 1.f;  a0.y = 1.f;  a1.x = 1.f; a1.y = 1.f; }
    {
      const int klo = (lane < 16) ? 0 : 2;
      const int col = lane & 3;              // cols 4..15 replicate -> ignored
      b.x = smem[klo * 4 + col];
      b.y = smem[(klo + 1) * 4 + col];
    }
    v8f c0 = {};
    v8f d0 = __builtin_amdgcn_wmma_f32_16x16x4_f32(false, a0, false, b,
                                                   (short)0, c0, false, false);
    v8f d1 = __builtin_amdgcn_wmma_f32_16x16x4_f32(false, a1, false, b,
                                                   (short)0, c0, false, false);
    // D layout: VGPR r -> M=r (lanes 0-15, N=lane) / M=8+r (lanes 16-31).
    // Scatter the 4 useful columns through LDS to per-pixel lanes.
    float* wbuf = &smem[32 + (tid >> 5) * 128];
    if (lane < 4) {
      #pragma unroll
      for (int r = 0; r < 8; ++r) {
        wbuf[r * 4 + lane]        = d0[r];   // pixels 0..7
        wbuf[(16 + r) * 4 + lane] = d1[r];   // pixels 16..23
      }
    } else if (lane >= 16 && lane < 20) {
      const int j = lane - 16;
      #pragma unroll
      for (int r = 0; r < 8; ++r) {
        wbuf[(8 + r) * 4 + j]  = d0[r];      // pixels 8..15
        wbuf[(24 + r) * 4 + j] = d1[r];      // pixels 24..31
      }
    }
    __syncthreads();
    const float4 w4 = *(const float4*)&wbuf[lane * 4];
    const float inw = 1.f / (w4.w + EPSF);
    const float cx = cam[0], cy = cam[1], cz = cam[2];
    ox = cx; oy = cy; oz = cz;
    dx = w4.x * inw - cx; dy = w4.y * inw - cy; dz = w4.z * inw - cz;
    const float nn = sqrtf(dx * dx + dy * dy + dz * dz) + EPSF;
    dx /= nn; dy /= nn; dz /= nn;
  } else {
    __syncthreads();                          // keep barriers uniform-ish
    const float* vi = &smem[16];
    float rx = -vi[8], ry = -vi[9], rz = -vi[10];          // -view_inv[2,:3]
    const float rn = sqrtf(rx * rx + ry * ry + rz * rz) + EPSF;
    rx /= rn; ry /= rn; rz /= rn;
    const float ex = bmaxx - bminx, ey = bmaxy - bminy, ez = bmaxz - bminz;
    const float diag = sqrtf(ex * ex + ey * ey + ez * ez);
    ox = (bminx + bmaxx) * 0.5f + ndcx * vi[0] * ex * 0.5f + ndcy * vi[4] * ey * 0.5f - rx * diag * 0.5f;
    oy = (bminy + bmaxy) * 0.5f + ndcx * vi[1] * ex * 0.5f + ndcy * vi[5] * ey * 0.5f - ry * diag * 0.5f;
    oz = (bminz + bmaxz) * 0.5f + ndcx * vi[2] * ex * 0.5f + ndcy * vi[6] * ey * 0.5f - rz * diag * 0.5f;
    dx = rx; dy = ry; dz = rz;
  }

  // ---- slab intersection (exactly mirrors reference incl. EPS quirks) -----
  const float ivx = 1.f / (dx + EPSF);
  const float ivy = 1.f / (dy + EPSF);
  const float ivz = 1.f / (dz + EPSF);
  const float t1x = (bminx - ox) * ivx, t2x = (bmaxx - ox) * ivx;
  const float t1y = (bminy - oy) * ivy, t2y = (bmaxy - oy) * ivy;
  const float t1z = (bminz - oz) * ivz, t2z = (bmaxz - oz) * ivz;
  float tn = fmaxf(fmaxf(fminf(t1x, t2x), fminf(t1y, t2y)), fminf(t1z, t2z));
  float tf = fminf(fminf(fmaxf(t1x, t2x), fmaxf(t1y, t2y)), fmaxf(t1z, t2z));
  tn = fmaxf(tn, 0.f);
  const bool valid = tf > tn;

  float acc = 0.f;
  if (valid) {
    const int RR = R * R;
    const float sx = (float)(R - 1) / (bmaxx - bminx + EPSF);
    const float sy = (float)(R - 1) / (bmaxy - bminy + EPSF);
    const float sz = (float)(R - 1) / (bmaxz - bminz + EPSF);
    const float bx = (ox - bminx) * sx, mx = dx * sx;
    const float by = (oy - bminy) * sy, my = dy * sy;
    const float bz = (oz - bminz) * sz, mz = dz * sz;
    const float dt63 = (tf - tn) * (1.f / (float)(NSAMP - 1));
    #pragma unroll 4
    for (int s = 0; s < NSAMP; ++s) {
      const float t = fmaf((float)s, dt63, tn);
      const float cxx = fmaf(t, mx, bx);
      const float cyy = fmaf(t, my, by);
      const float czz = fmaf(t, mz, bz);
      const float fx0 = floorf(cxx), fy0 = floorf(cyy), fz0 = floorf(czz);
      const int ix = (int)fx0, iy = (int)fy0, iz = (int)fz0;
      const float ax = cxx - fx0, ay = cyy - fy0, az = czz - fz0;
      // zero-padding semantics: mask per-axis weights by index validity
      const float wx0 = ((unsigned)ix       < (unsigned)R) ? (1.f - ax) : 0.f;
      const float wx1 = ((unsigned)(ix + 1) < (unsigned)R) ? ax         : 0.f;
      const float wy0 = ((unsigned)iy       < (unsigned)R) ? (1.f - ay) : 0.f;
      const float wy1 = ((unsigned)(iy + 1) < (unsigned)R) ? ay         : 0.f;
      const float wz0 = ((unsigned)iz       < (unsigned)R) ? (1.f - az) : 0.f;
      const float wz1 = ((unsigned)(iz + 1) < (unsigned)R) ? az         : 0.f;
      const int X0 = min(max(ix, 0), R - 1),     X1 = min(max(ix + 1, 0), R - 1);
      const int Y0 = min(max(iy, 0), R - 1),     Y1 = min(max(iy + 1, 0), R - 1);
      const int Z0 = min(max(iz, 0), R - 1),     Z1 = min(max(iz + 1, 0), R - 1);
      const int z0o = Z0 * RR, z1o = Z1 * RR, y0o = Y0 * R, y1o = Y1 * R;
      const float v000 = vol[z0o + y0o + X0], v001 = vol[z0o + y0o + X1];
      const float v010 = vol[z0o + y1o + X0], v011 = vol[z0o + y1o + X1];
      const float v100 = vol[z1o + y0o + X0], v101 = vol[z1o + y0o + X1];
      const float v110 = vol[z1o + y1o + X0], v111 = vol[z1o + y1o + X1];
      const float wy0z0 = wy0 * wz0, wy1z0 = wy1 * wz0;
      const float wy0z1 = wy0 * wz1, wy1z1 = wy1 * wz1;
      acc += wx0 * (v000 * wy0z0 + v010 * wy1z0 + v100 * wy0z1 + v110 * wy1z1)
           + wx1 * (v001 * wy0z0 + v011 * wy1z0 + v101 * wy0z1 + v111 * wy1z1);
      // software prefetch of next sample's low corner (L2 latency hiding)
      const float t2 = t + dt63;
      const int nx = min(max((int)floorf(fmaf(t2, mx, bx)), 0), R - 1);
      const int ny = min(max((int)floorf(fmaf(t2, my, by)), 0), R - 1);
      const int nz = min(max((int)floorf(fmaf(t2, mz, bz)), 0), R - 1);
      __builtin_prefetch(vol + nz * RR + ny * R + nx, 0, 0);
    }
  }
  if (px < W && py < H)
    out[py * W + px] = valid ? acc * ((tf - tn) * (1.f / (float)NSAMP)) : 0.f;
}

extern "C" void kernel_launch(void* const* d_in, const int* in_sizes, int n_in,
                              void* d_out, int out_size, void* d_ws, size_t ws_size,
                              hipStream_t stream) {
  const float* vol   = (const float*)d_in[0];
  const float* wvt   = (const float*)d_in[1];
  const float* proj  = (const float*)d_in[2];
  const float* cam   = (const float*)d_in[3];
  const float* bbox  = (const float*)d_in[4];
  const int*   Hp    = (const int*)d_in[5];
  const int*   Wp    = (const int*)d_in[6];
  const int*   modep = (const int*)d_in[7];
  float* ws  = (float*)d_ws;
  float* out = (float*)d_out;

  int R = (int)(cbrt((double)in_sizes[0]) + 0.5);           // 128
  int side = (int)(sqrt((double)out_size) + 0.5);           // 512
  if (side < 1) side = 1;

  dsv_setup_kernel<<<1, 32, 0, stream>>>(wvt, proj, ws);
  dim3 grid((side + 15) / 16, (side + 15) / 16);
  dsv_render_kernel<<<grid, 256, 0, stream>>>(vol, Hp, Wp, modep, cam, bbox, ws,
                                              out, R);
}